// MultiViewMBPredictor_12043088298471
// MI455X (gfx1250) — compile-verified
//
#include <hip/hip_runtime.h>
#include <math.h>

// ---------------------------------------------------------------------------
// Problem sizes (fixed by the reference)
// ---------------------------------------------------------------------------
#define Bsz 16
#define Ssz 4096
#define Fsz 512
#define NC  10
#define HID 64
#define OUTD 128
#define NROW (Bsz * Fsz)          // 8192 (b,f) columns / MLP rows

// Workspace layout (floats). Total ~770 KB.
#define ACC_STRIDE   16
#define ACC_FLOATS   (NROW * ACC_STRIDE)            // per-column accumulators
#define CNT_OFF      ACC_FLOATS
#define CNT_FLOATS   (Bsz * 16)                     // class counts per batch
#define STATS_OFF    (CNT_OFF + CNT_FLOATS)
#define STATS_FLOATS (NROW * 8)                     // stats padded K=6 -> 8
#define WS_FLOATS    (STATS_OFF + STATS_FLOATS)

typedef float v2f __attribute__((ext_vector_type(2)));
typedef float v8f __attribute__((ext_vector_type(8)));
typedef unsigned int u32x4 __attribute__((ext_vector_type(4)));
typedef int i32x4 __attribute__((ext_vector_type(4)));
typedef int i32x8 __attribute__((ext_vector_type(8)));

// ---------------------------------------------------------------------------
// Kernel 0: zero the workspace (ws is poisoned once; must be re-inited
// deterministically every kernel_launch call)
// ---------------------------------------------------------------------------
__global__ void zero_ws_kernel(float* ws, int n) {
  int i = blockIdx.x * blockDim.x + threadIdx.x;
  int stride = gridDim.x * blockDim.x;
  for (; i < n; i += stride) ws[i] = 0.0f;
}

// ---------------------------------------------------------------------------
// Kernel 1: per-batch class histogram (counts[b][c])
// ---------------------------------------------------------------------------
__global__ void count_classes_kernel(const int* __restrict__ y, float* ws) {
  __shared__ int hist[NC];
  int b = blockIdx.x;
  if (threadIdx.x < NC) hist[threadIdx.x] = 0;
  __syncthreads();
  for (int s = threadIdx.x; s < Ssz; s += blockDim.x) {
    int c = y[b * Ssz + s];
    if (c >= 0 && c < NC) atomicAdd(&hist[c], 1);
  }
  __syncthreads();
  if (threadIdx.x < NC) ws[CNT_OFF + b * 16 + threadIdx.x] = (float)hist[threadIdx.x];
}

// ---------------------------------------------------------------------------
// Kernel 2: coalesced streaming accumulation over X (bandwidth leg:
// 128 MB @ 23.3 TB/s ~= 5.5 us floor). Block = 512 threads = one thread per
// feature; each loop iteration reads one fully-contiguous 2KB row.
// ---------------------------------------------------------------------------
#define CHUNKS 8
#define ROWS_PER_CHUNK (Ssz / CHUNKS)   // 512

__global__ __launch_bounds__(Fsz) void accum_stats_kernel(
    const float* __restrict__ X, const int* __restrict__ y, float* ws) {
  int b     = blockIdx.x / CHUNKS;
  int chunk = blockIdx.x % CHUNKS;
  int f     = threadIdx.x;

  float sum = 0.0f, sumsq = 0.0f, sumabs = 0.0f, maxabs = 0.0f, nancnt = 0.0f;
  float csum[NC] = {};

  const float* Xb = X + (size_t)b * Ssz * Fsz;
  const int*   yb = y + b * Ssz;

  for (int i = 0; i < ROWS_PER_CHUNK; ++i) {
    int s = chunk * ROWS_PER_CHUNK + i;
    if (i + 16 < ROWS_PER_CHUNK)   // keep the load pipe full (global_prefetch)
      __builtin_prefetch(&Xb[(size_t)(s + 16) * Fsz + f], 0, 0);
    float x = Xb[(size_t)s * Fsz + f];
    bool nan = isnan(x);
    float xc = nan ? 0.0f : x;
    nancnt += nan ? 1.0f : 0.0f;
    sum    += xc;
    sumsq  += xc * xc;
    float a = fabsf(xc);
    sumabs += a;
    maxabs  = fmaxf(maxabs, a);
    int c = yb[s];   // uniform across the block
#pragma unroll
    for (int j = 0; j < NC; ++j) csum[j] += (j == c) ? xc : 0.0f;
  }

  float* acc = ws + (size_t)(b * Fsz + f) * ACC_STRIDE;
  atomicAdd(&acc[0], sum);
  atomicAdd(&acc[1], sumsq);
  atomicAdd(&acc[2], sumabs);
  // max|x| >= 0 so int-bit comparison is order-preserving; init bits are 0
  atomicMax((int*)&acc[3], __float_as_int(maxabs));
  atomicAdd(&acc[4], nancnt);
#pragma unroll
  for (int j = 0; j < NC; ++j) atomicAdd(&acc[5 + j], csum[j]);
}

// ---------------------------------------------------------------------------
// Kernel 3: per-column bitonic sort in LDS (16 KB), unique count, and stat
// finalization. One block per (b,f) column; X is L2-resident (128MB < 192MB).
// The strided column gather (4096 x 4B, stride 2KB) is done by the Tensor
// Data Mover: 2D D# with tile_dim0=1, tile_dim1=4096, dim0_stride=512 packs
// the column contiguously into LDS (descriptor per CDNA5 ISA).
// ---------------------------------------------------------------------------
__global__ __launch_bounds__(256) void sort_finalize_kernel(
    const float* __restrict__ X, float* ws) {
  __shared__ float xs[Ssz];
  __shared__ int   red[256];

  int col = blockIdx.x;              // 0..NROW-1
  int b = col / Fsz;
  int f = col % Fsz;
  const float* Xcol = X + (size_t)b * Ssz * Fsz + f;

#if __has_builtin(__builtin_amdgcn_tensor_load_to_lds) && \
    __has_builtin(__builtin_amdgcn_s_wait_tensorcnt)
  if (threadIdx.x < 32) {            // one TDM issue per block (TDM ignores EXEC)
    unsigned long long ga = (unsigned long long)(uintptr_t)Xcol;
    unsigned lds = (unsigned)(uintptr_t)&xs[0];
    u32x4 g0;
    g0.x = 1u;                                    // count=1, user mode, no gather
    g0.y = lds;                                   // lds_addr [63:32]
    g0.z = (unsigned)(ga & 0xFFFFFFFFu);          // global_addr [95:64]
    g0.w = (unsigned)((ga >> 32) & 0x01FFFFFFu)   // global_addr [120:96]
         | 0x80000000u;                           // type=2 ("image") [127:126]
    i32x8 g1;
    g1[0] = (int)(2u << 16);        // data_size=4B; workgroup_mask=0; no flags
    g1[1] = (int)(1u << 16);        // tensor_dim0 = 1
    g1[2] = (int)(4096u << 16);     // tensor_dim1 = 4096 (low 16 in [31:16])
    g1[3] = (int)(1u << 16);        // tile_dim0 = 1
    g1[4] = (int)4096;              // tile_dim1 = 4096, tile_dim2 = 0
    g1[5] = (int)Fsz;               // tensor_dim0_stride = 512 elements
    g1[6] = 0;                      // dim0_stride hi, dim1_stride lo
    g1[7] = 0;                      // dim1_stride hi
    i32x4 gz4 = {0, 0, 0, 0};       // 2D tensor: groups 2/3 unused
    i32x8 gz8 = {0, 0, 0, 0, 0, 0, 0, 0};  // trailing group slot (unused, count=1)
    __builtin_amdgcn_tensor_load_to_lds(g0, g1, gz4, gz4, gz8, 0);
    __builtin_amdgcn_s_wait_tensorcnt(0);
  }
  __syncthreads();
  // nan_to_num pass (reference sorts Xc = nan->0)
  for (int s = threadIdx.x; s < Ssz; s += 256) {
    float v = xs[s];
    if (isnan(v)) xs[s] = 0.0f;
  }
#else
  for (int s = threadIdx.x; s < Ssz; s += 256) {
    float x = Xcol[(size_t)s * Fsz];
    xs[s] = isnan(x) ? 0.0f : x;
  }
#endif
  __syncthreads();

  // bitonic sort, ascending
  for (int k = 2; k <= Ssz; k <<= 1) {
    for (int j = k >> 1; j > 0; j >>= 1) {
      for (int i = threadIdx.x; i < Ssz; i += 256) {
        int p = i ^ j;
        if (p > i) {
          bool up = ((i & k) == 0);
          float a = xs[i], c = xs[p];
          if ((a > c) == up) { xs[i] = c; xs[p] = a; }
        }
      }
      __syncthreads();
    }
  }

  // count adjacent-distinct values
  int loc = 0;
  for (int i = threadIdx.x; i < Ssz; i += 256)
    if (i > 0) loc += (xs[i] != xs[i - 1]) ? 1 : 0;
  red[threadIdx.x] = loc;
  __syncthreads();
  for (int st = 128; st > 0; st >>= 1) {
    if (threadIdx.x < st) red[threadIdx.x] += red[threadIdx.x + st];
    __syncthreads();
  }

  if (threadIdx.x == 0) {
    const float S = (float)Ssz;
    const float* acc = ws + (size_t)col * ACC_STRIDE;
    const float* cnt = ws + CNT_OFF + b * 16;

    float n_unique  = 1.0f + (float)red[0];
    float sum       = acc[0];
    float sumsq     = acc[1];
    float sumabs    = acc[2];
    float maxabs    = acc[3];               // float bits written via int atomicMax
    float nancnt    = acc[4];

    float gmean    = sum / S;
    float variance = sumsq / S - gmean * gmean;        // biased, matches jnp.var
    float mean_abs = sumabs / S;
    float missing  = nancnt / S;
    float uratio   = n_unique / S;

    float between = 0.0f;
#pragma unroll
    for (int c = 0; c < NC; ++c) {
      float cn = cnt[c];
      float cm = acc[5 + c] / fmaxf(cn, 1.0f);
      float d  = cm - gmean;
      between += cn * d * d;
    }
    between /= S;                                       // counts.sum == S
    float tscore = between / fmaxf(variance, 1e-6f);

    float st6[6] = {tscore, missing, uratio, variance, mean_abs, maxabs};
    float* o = ws + STATS_OFF + (size_t)col * 8;
#pragma unroll
    for (int i = 0; i < 6; ++i) o[i] = isfinite(st6[i]) ? st6[i] : 0.0f;
    // o[6], o[7] stay zero (K padding for WMMA)
  }
}

// ---------------------------------------------------------------------------
// Kernel 4: MLP via V_WMMA_F32_16X16X4_F32.
// Block = 128 threads = 4 waves; each wave owns one 16-row tile.
// GEMM1: [16x8(K=6 padded)] x [8x64]  -> bias + exact GELU -> LDS
// GEMM2: [16x64] x [64x128]           -> + b2 -> out
// w1 staged zero-padded in LDS so B loads are unconditional (no EXEC churn
// around the WMMAs). Hidden staged through padded LDS (D-layout -> A-layout).
// ---------------------------------------------------------------------------
__global__ __launch_bounds__(128) void mlp_wmma_kernel(
    const float* __restrict__ ws, const float* __restrict__ w1,
    const float* __restrict__ b1, const float* __restrict__ w2,
    const float* __restrict__ b2, float* __restrict__ out) {
  __shared__ float hid[4][16][HID + 1];   // +1 pad: kills LDS bank conflicts
  __shared__ float w1s[8 * HID];          // w1 zero-padded K=6 -> 8

  const int wave = threadIdx.x >> 5;
  const int lane = threadIdx.x & 31;
  const int m0   = (blockIdx.x * 4 + wave) * 16;   // global row-tile base
  const int mrow = lane & 15;                      // A-layout row
  const int koff = (lane >> 4) * 2;                // A/B-layout K sub-offset
  const int ncol = lane & 15;                      // B/D-layout column
  const int drow = 8 * (lane >> 4);                // D-layout row base

  for (int i = threadIdx.x; i < 8 * HID; i += 128)
    w1s[i] = (i < 6 * HID) ? w1[i] : 0.0f;

  const float* stats = ws + STATS_OFF;

  // ---- GEMM1 A: stats tile (K padded to 8), two K-blocks of 4 ----
  v2f A1[2];
#pragma unroll
  for (int kb = 0; kb < 2; ++kb) {
    int k0 = kb * 4 + koff;
    A1[kb].x = stats[(size_t)(m0 + mrow) * 8 + k0];
    A1[kb].y = stats[(size_t)(m0 + mrow) * 8 + k0 + 1];
  }
  __syncthreads();   // w1s ready

#pragma unroll
  for (int nt = 0; nt < 4; ++nt) {                 // 4 x 16 = 64 hidden cols
    int n = nt * 16 + ncol;
    v8f acc = {};
#pragma unroll
    for (int kb = 0; kb < 2; ++kb) {
      int k0 = kb * 4 + koff;
      v2f Bm;
      Bm.x = w1s[k0 * HID + n];
      Bm.y = w1s[(k0 + 1) * HID + n];
      acc = __builtin_amdgcn_wmma_f32_16x16x4_f32(
          false, A1[kb], false, Bm, (short)0, acc, false, false);
    }
#pragma unroll
    for (int v = 0; v < 8; ++v) {                  // D-layout: row = v + drow
      float h = acc[v] + b1[n];
      float g = 0.5f * h * (1.0f + erff(h * 0.70710678118654752f));  // exact GELU
      hid[wave][v + drow][n] = g;
    }
  }
  __syncthreads();

  // ---- GEMM2: A = hidden (K = 64), 16 K-blocks held in registers ----
  v2f A2[16];
#pragma unroll
  for (int kb = 0; kb < 16; ++kb) {
    int k0 = kb * 4 + koff;
    A2[kb].x = hid[wave][mrow][k0];
    A2[kb].y = hid[wave][mrow][k0 + 1];
  }

#pragma unroll
  for (int ot = 0; ot < 8; ++ot) {                 // 8 x 16 = 128 output cols
    int n = ot * 16 + ncol;
    v8f acc = {};
#pragma unroll
    for (int kb = 0; kb < 16; ++kb) {
      int k0 = kb * 4 + koff;
      v2f Bm;
      Bm.x = w2[k0 * OUTD + n];
      Bm.y = w2[(k0 + 1) * OUTD + n];
      acc = __builtin_amdgcn_wmma_f32_16x16x4_f32(
          false, A2[kb], false, Bm, (short)0, acc, false, false);
    }
#pragma unroll
    for (int v = 0; v < 8; ++v) {
      int r = m0 + v + drow;
      out[(size_t)r * OUTD + n] = acc[v] + b2[n];
    }
  }
}

// ---------------------------------------------------------------------------
// Launch
// ---------------------------------------------------------------------------
extern "C" void kernel_launch(void* const* d_in, const int* in_sizes, int n_in,
                              void* d_out, int out_size, void* d_ws, size_t ws_size,
                              hipStream_t stream) {
  const float* X  = (const float*)d_in[0];
  const int*   y  = (const int*)  d_in[1];
  const float* w1 = (const float*)d_in[2];
  const float* b1 = (const float*)d_in[3];
  const float* w2 = (const float*)d_in[4];
  const float* b2 = (const float*)d_in[5];
  float* out = (float*)d_out;
  float* ws  = (float*)d_ws;   // needs WS_FLOATS*4 ~ 770 KB

  zero_ws_kernel      <<<256, 256, 0, stream>>>(ws, WS_FLOATS);
  count_classes_kernel<<<Bsz, 256, 0, stream>>>(y, ws);
  accum_stats_kernel  <<<Bsz * CHUNKS, Fsz, 0, stream>>>(X, y, ws);
  sort_finalize_kernel<<<NROW, 256, 0, stream>>>(X, ws);
  mlp_wmma_kernel     <<<NROW / 64, 128, 0, stream>>>(ws, w1, b1, w2, b2, out);
}